// TokenSparse_57449482551493
// MI455X (gfx1250) — compile-verified
//
#include <hip/hip_runtime.h>
#include <hip/hip_bf16.h>
#include <stdint.h>

// Problem constants (from reference): tokens (64,4096,512) f32
#define BB 64
#define NN 4096
#define CC 512
#define KK 2458          // ceil(4096*0.6)
#define NK (NN - KK)     // 1638
#define ROWS_PER_BLOCK 8 // gather pipelining depth (16KB LDS of 320KB)

typedef __attribute__((ext_vector_type(2))) float v2f;
typedef __attribute__((ext_vector_type(8))) float v8f;

// Monotonic float<->uint mapping for sortable keys
__device__ __forceinline__ unsigned enc_f32(float f) {
    unsigned u = __float_as_uint(f);
    return (u >> 31) ? ~u : (u ^ 0x80000000u);
}
__device__ __forceinline__ float dec_f32(unsigned v) {
    return (v >> 31) ? __uint_as_float(v ^ 0x80000000u) : __uint_as_float(~v);
}

// ---------------------------------------------------------------------------
// Kernel 1: per-batch bitonic sort of (score,idx) keys in LDS; emits mask,
// keep indices, non-keep indices, unnormalized softmax weights e, and sum(e).
// One workgroup (1024 threads = 32 wave32s) per batch; 32KB LDS keys.
// ---------------------------------------------------------------------------
__global__ __launch_bounds__(1024) void sort_kernel(
    const float* __restrict__ ax, const float* __restrict__ ay,
    float* __restrict__ mask_out, int* __restrict__ keep_idx,
    int* __restrict__ nk_idx, float* __restrict__ e_out,
    float* __restrict__ sum_out)
{
    __shared__ unsigned long long keys[NN];
    __shared__ float red[1024];
    const int b = blockIdx.x;
    const int tid = threadIdx.x;

    // Build keys: sort ascending by (~enc(score))<<32 | idx
    //  -> descending score, ties broken by ascending index (stable argsort)
    for (int i = tid; i < NN; i += 1024) {
        float s = ax[(size_t)b * NN + i] + ay[(size_t)b * NN + i];
        unsigned u = enc_f32(s);
        keys[i] = ((unsigned long long)(~u) << 32) | (unsigned)i;
    }
    __syncthreads();

    // Bitonic sort, 4096 elements, 1024 threads (each owns 4 slots)
    for (int k = 2; k <= NN; k <<= 1) {
        for (int j = k >> 1; j > 0; j >>= 1) {
            for (int t = tid; t < NN; t += 1024) {
                int ixj = t ^ j;
                if (ixj > t) {
                    bool up = ((t & k) == 0);
                    unsigned long long a = keys[t];
                    unsigned long long c = keys[ixj];
                    if ((a > c) == up) { keys[t] = c; keys[ixj] = a; }
                }
            }
            __syncthreads();
        }
    }

    // Softmax max = score at sorted position KK (largest non-kept score)
    const float m = dec_f32(~(unsigned)(keys[KK] >> 32));

    float lsum = 0.0f;
    for (int i = tid; i < NN; i += 1024) {
        unsigned long long key = keys[i];
        int idx = (int)(unsigned)(key & 0xffffffffu);
        mask_out[(size_t)b * NN + idx] = (i < KK) ? 1.0f : 0.0f;
        if (i < KK) {
            keep_idx[(size_t)b * KK + i] = idx;
        } else {
            float s  = dec_f32(~(unsigned)(key >> 32));
            float ev = __expf(s - m);
            e_out [(size_t)b * NK + (i - KK)] = ev;
            nk_idx[(size_t)b * NK + (i - KK)] = idx;
            lsum += ev;
        }
    }
    red[tid] = lsum;
    __syncthreads();
    for (int s = 512; s > 0; s >>= 1) {
        if (tid < s) red[tid] += red[tid + s];
        __syncthreads();
    }
    if (tid == 0) sum_out[b] = red[0];
}

// ---------------------------------------------------------------------------
// Kernel 2: gather select_tokens rows via the CDNA5 async global<->LDS path,
// software-pipelined: 8 rows per 128-thread block. Phase 1 issues 8 async
// loads back-to-back (8 outstanding on ASYNCcnt per wave), one
// s_wait_asynccnt 0, then 8 async stores. Non-temporal hints: kept rows and
// the select_tokens stream have zero reuse, keep them out of the 192MB L2.
// Dynamic LDS (16KB) starts at LDS offset 0 (no static LDS in this kernel).
// ---------------------------------------------------------------------------
__global__ __launch_bounds__(128) void gather_kernel(
    const float* __restrict__ tokens, const int* __restrict__ keep_idx,
    float* __restrict__ sel)
{
    extern __shared__ char smem_dyn[];
    (void)smem_dyn;
    const int b  = blockIdx.y;
    const int r0 = blockIdx.x * ROWS_PER_BLOCK;
    const unsigned tid = threadIdx.x;

    // Phase 1: global -> LDS, 8 rows in flight
    #pragma unroll
    for (int i = 0; i < ROWS_PER_BLOCK; ++i) {
        int r = r0 + i;
        if (r < KK) {                                   // uniform per block
            int src = keep_idx[(size_t)b * KK + r];     // broadcast load
            unsigned lds_off = (unsigned)i * 2048u + tid * 16u;
            unsigned long long sp = (unsigned long long)(uintptr_t)
                (tokens + ((size_t)b * NN + (size_t)src) * CC + (size_t)tid * 4);
            asm volatile("global_load_async_to_lds_b128 %0, %1, off th:TH_LOAD_NT"
                         :: "v"(lds_off), "v"(sp) : "memory");
        }
    }
    asm volatile("s_wait_asynccnt 0" ::: "memory");

    // Phase 2: LDS -> global, 8 rows in flight (drained by the final s_wait_asynccnt 0)
    #pragma unroll
    for (int i = 0; i < ROWS_PER_BLOCK; ++i) {
        int r = r0 + i;
        if (r < KK) {
            unsigned lds_off = (unsigned)i * 2048u + tid * 16u;
            unsigned long long dp = (unsigned long long)(uintptr_t)
                (sel + ((size_t)b * KK + (size_t)r) * CC + (size_t)tid * 4);
            asm volatile("global_store_async_from_lds_b128 %0, %1, off th:TH_STORE_NT"
                         :: "v"(dp), "v"(lds_off) : "memory");
        }
    }
    asm volatile("s_wait_asynccnt 0" ::: "memory");
}

// ---------------------------------------------------------------------------
// Kernel 3: extra_token[b,c] = (1/sum) * sum_j e_j * tokens[b, nk_j, c]
// as a 1 x NK x 512 GEMV on v_wmma_f32_16x16x4_f32, f32 end to end.
// 32 waves per batch; wave w owns channels [16w, 16w+16). Per step, 4 token
// rows enter the B (4x16) tile; softmax weights sit in row 0 of the A (16x4)
// tile. Layouts per ISA 7.12.2:
//   A 16x4 f32 : lane L (<16): v0=A[L][0], v1=A[L][1]; lane L+16: K=2,3
//   B 4x16 f32 : v0 = rows {K=0 (lanes 0-15), K=2 (lanes 16-31)},
//                v1 = rows {K=1, K=3}   (mirrors the A K-split)
//   D 16x16 f32: v0, lanes 0-15 = row M=0 -> our result row.
// ---------------------------------------------------------------------------
__global__ __launch_bounds__(1024) void extra_kernel(
    const float* __restrict__ tokens, const int* __restrict__ nk_idx,
    const float* __restrict__ e, const float* __restrict__ sum,
    float* __restrict__ extra)
{
    const int b    = blockIdx.x;
    const int lane = threadIdx.x & 31;
    const int wv   = threadIdx.x >> 5;    // 0..31
    const int c0   = wv * 16;
    const int l    = lane & 15;
    const int half = lane >> 4;           // 0: K=0,1  1: K=2,3

    const float* eb = e      + (size_t)b * NK;
    const int*   nb = nk_idx + (size_t)b * NK;
    const float* tb = tokens + (size_t)b * NN * CC;

    v8f acc = {};
    for (int j = 0; j < NK; j += 4) {
        int r0 = j + half * 2;
        int r1 = r0 + 1;
        float f0 = (r0 < NK) ? 1.0f : 0.0f;   // tail masking (NK%4 == 2)
        float f1 = (r1 < NK) ? 1.0f : 0.0f;
        int cr0 = (r0 < NK) ? r0 : (NK - 1);
        int cr1 = (r1 < NK) ? r1 : (NK - 1);

        // A tile: only row M=0 is non-zero (the weights)
        float ev0 = eb[cr0] * f0;
        float ev1 = eb[cr1] * f1;
        v2f a;
        a.x = (l == 0) ? ev0 : 0.0f;
        a.y = (l == 0) ? ev1 : 0.0f;

        // B tile: 4 gathered token rows, 16 channels wide
        v2f bt;
        bt.x = tb[(size_t)nb[cr0] * CC + c0 + l] * f0;
        bt.y = tb[(size_t)nb[cr1] * CC + c0 + l] * f1;

        acc = __builtin_amdgcn_wmma_f32_16x16x4_f32(
            /*neg_a=*/false, a, /*neg_b=*/false, bt,
            /*c_mod=*/(short)0, acc, /*reuse_a=*/false, /*reuse_b=*/false);
    }

    // Row M=0 of D lives in acc[0] on lanes 0-15
    if (half == 0) {
        extra[(size_t)b * CC + c0 + l] = acc[0] / sum[b];
    }
}

// ---------------------------------------------------------------------------
extern "C" void kernel_launch(void* const* d_in, const int* in_sizes, int n_in,
                              void* d_out, int out_size, void* d_ws, size_t ws_size,
                              hipStream_t stream) {
    const float* tokens = (const float*)d_in[0];
    const float* ax     = (const float*)d_in[1];
    const float* ay     = (const float*)d_in[2];

    float* out   = (float*)d_out;
    float* sel   = out;                                   // B*K*C
    float* extra = out + (size_t)BB * KK * CC;            // B*C
    float* mask  = extra + (size_t)BB * CC;               // B*N

    char* ws = (char*)d_ws;
    int*   keep_idx = (int*)ws;
    int*   nk_idxp  = (int*)(ws + sizeof(int) * (size_t)BB * KK);
    float* e_ws     = (float*)(ws + sizeof(int) * (size_t)BB * (KK + NK));
    float* sum_ws   = (float*)(ws + sizeof(int) * (size_t)BB * (KK + NK)
                                  + sizeof(float) * (size_t)BB * NK);

    sort_kernel<<<BB, 1024, 0, stream>>>(ax, ay, mask, keep_idx, nk_idxp, e_ws, sum_ws);

    const int gx = (KK + ROWS_PER_BLOCK - 1) / ROWS_PER_BLOCK;   // 308
    gather_kernel<<<dim3(gx, BB), 128, ROWS_PER_BLOCK * 128 * 16, stream>>>(
        tokens, keep_idx, sel);

    extra_kernel<<<BB, 1024, 0, stream>>>(tokens, nk_idxp, e_ws, sum_ws, extra);
}